// AttnDecoderRNN_new_4750233829836
// MI455X (gfx1250) — compile-verified
//
#include <hip/hip_runtime.h>
#include <math.h>

typedef __attribute__((ext_vector_type(2))) float v2f;
typedef __attribute__((ext_vector_type(4))) float v4f;
typedef __attribute__((ext_vector_type(8))) float v8f;

#define HIDDEN  32
#define VOCAB   8192
#define MAX_LEN 100
#define BASKET  20
#define KSPLIT  8
#define LINE    32                       // floats per skip unit (one 128B line)
#define NLINES  (VOCAB / LINE)           // 256 lines -> 8 mask dwords
// workspace float offsets
#define WS_HNEW  VOCAB                   // [8192,8224): h_new
#define WS_BLEND (VOCAB + HIDDEN)        // [8224,16416): blended logits
#define WS_MASK  (VOCAB + HIDDEN + VOCAB) // [16416,+8 dwords): history line mask

// ---------------------------------------------------------------------------
// K1: all the tiny sequential work (emb sum, attention, softmax(100),
// combine, GRU step) in a single block, plus zero-init of the c5 accumulator
// and the per-cacheline nonzero bitmask of the sparse history vector.
// ---------------------------------------------------------------------------
__global__ __launch_bounds__(128) void k1_small(
    const int* __restrict__ ids, const float* __restrict__ hidden,
    const float* __restrict__ enc, const float* __restrict__ attn_W,
    const float* __restrict__ attn_b, const float* __restrict__ comb_W,
    const float* __restrict__ comb_b, const float* __restrict__ W_ih,
    const float* __restrict__ W_hh, const float* __restrict__ b_ih,
    const float* __restrict__ b_hh, const float* __restrict__ emb,
    const float* __restrict__ hist,
    float* __restrict__ d_out, float* __restrict__ ws)
{
  __shared__ float s_cat[2 * HIDDEN];   // [emb_sum | h0] then [emb_sum | attn_applied]
  __shared__ float s_attn[MAX_LEN];
  __shared__ float s_red[128];
  __shared__ float s_x[HIDDEN];
  __shared__ float s_gx[3 * HIDDEN];
  __shared__ float s_gh[3 * HIDDEN];
  __shared__ unsigned char s_line[NLINES];
  const int t = threadIdx.x;

  // zero c5 GEMV accumulator (K2 atomically accumulates into it)
  for (int i = t; i < VOCAB; i += 128) ws[i] = 0.f;

  // per-128B-line nonzero flags for history
  for (int line = t; line < NLINES; line += 128) {
    const float* p = hist + line * LINE;
    bool nz = false;
#pragma unroll
    for (int q = 0; q < LINE; q += 4) {
      const v4f h = *(const v4f*)(p + q);
      nz |= (h.x != 0.f) | (h.y != 0.f) | (h.z != 0.f) | (h.w != 0.f);
    }
    s_line[line] = nz ? 1 : 0;
  }

  // emb_sum over basket + concat with h0
  if (t < HIDDEN) {
    float s = 0.f;
    for (int k = 0; k < BASKET; ++k) s += emb[ids[k] * HIDDEN + t];
    s_cat[t] = s;
    s_cat[HIDDEN + t] = hidden[t];
  }
  __syncthreads();

  // pack line flags into 8 mask dwords
  if (t < NLINES / 32) {
    unsigned w = 0;
    for (int b = 0; b < 32; ++b) w |= (unsigned)s_line[t * 32 + b] << b;
    ((unsigned*)(ws + WS_MASK))[t] = w;
  }

  // attention scores (100 x 64 GEMV)
  float sc = -INFINITY;
  if (t < MAX_LEN) {
    float a = attn_b[t];
    for (int k = 0; k < 2 * HIDDEN; ++k) a += attn_W[t * 2 * HIDDEN + k] * s_cat[k];
    sc = a;
  }
  s_red[t] = sc;
  __syncthreads();
  for (int off = 64; off > 0; off >>= 1) {
    if (t < off) s_red[t] = fmaxf(s_red[t], s_red[t + off]);
    __syncthreads();
  }
  const float mx = s_red[0];
  __syncthreads();
  const float ex = (t < MAX_LEN) ? expf(sc - mx) : 0.f;
  s_red[t] = ex;
  __syncthreads();
  for (int off = 64; off > 0; off >>= 1) {
    if (t < off) s_red[t] += s_red[t + off];
    __syncthreads();
  }
  const float inv = 1.f / s_red[0];
  __syncthreads();
  if (t < MAX_LEN) {
    const float w = ex * inv;
    s_attn[t] = w;
    d_out[VOCAB + HIDDEN + t] = w;   // attn_weights output
  }
  __syncthreads();

  // attn_applied = attn_weights @ encoder_outputs ; build cat2
  if (t < HIDDEN) {
    float a = 0.f;
    for (int l = 0; l < MAX_LEN; ++l) a += s_attn[l] * enc[l * HIDDEN + t];
    s_cat[HIDDEN + t] = a;           // s_cat[0:32] still holds emb_sum
  }
  __syncthreads();

  // combine + relu
  if (t < HIDDEN) {
    float a = comb_b[t];
    for (int k = 0; k < 2 * HIDDEN; ++k) a += comb_W[t * 2 * HIDDEN + k] * s_cat[k];
    s_x[t] = fmaxf(a, 0.f);
  }
  __syncthreads();

  // GRU gate pre-activations
  if (t < 3 * HIDDEN) {
    float a = b_ih[t], b = b_hh[t];
    for (int k = 0; k < HIDDEN; ++k) {
      a += W_ih[t * HIDDEN + k] * s_x[k];
      b += W_hh[t * HIDDEN + k] * hidden[k];
    }
    s_gx[t] = a;
    s_gh[t] = b;
  }
  __syncthreads();

  // GRU update
  if (t < HIDDEN) {
    const float r = 1.f / (1.f + expf(-(s_gx[t] + s_gh[t])));
    const float z = 1.f / (1.f + expf(-(s_gx[HIDDEN + t] + s_gh[HIDDEN + t])));
    const float n = tanhf(s_gx[2 * HIDDEN + t] + r * s_gh[2 * HIDDEN + t]);
    const float hn = (1.f - z) * n + z * hidden[t];
    d_out[VOCAB + t] = hn;           // h_new output
    ws[WS_HNEW + t]  = hn;           // for K3
  }
}

// ---------------------------------------------------------------------------
// Helper: two V_WMMA_F32_16X16X4_F32 steps covering 8 K-columns.
// Per lane, `a` is a b128 of 4 consecutive A-columns:
//   lanes 0-15  : cols j..j+3   (feed K-slots 0,1 of each WMMA)
//   lanes 16-31 : cols j+4..j+7 (feed K-slots 2,3 of each WMMA)
// h0 = vector[j..j+3], h1 = vector[j+4..j+7] (uniform), replicated across all
// N-columns so every column of D carries the GEMV result.
// ---------------------------------------------------------------------------
__device__ __forceinline__ v8f wmma_gemv8(v8f acc, v4f a, v4f h0, v4f h1, bool lo)
{
  v2f a0, a1, b0, b1;
  a0.x = a.x; a0.y = a.y;            // cols j,j+1   | j+4,j+5
  a1.x = a.z; a1.y = a.w;            // cols j+2,j+3 | j+6,j+7
  b0.x = lo ? h0.x : h1.x;           // K0=h[j],   K2=h[j+4]
  b0.y = lo ? h0.y : h1.y;           // K1=h[j+1], K3=h[j+5]
  b1.x = lo ? h0.z : h1.z;           // K0=h[j+2], K2=h[j+6]
  b1.y = lo ? h0.w : h1.w;           // K1=h[j+3], K3=h[j+7]
  acc = __builtin_amdgcn_wmma_f32_16x16x4_f32(false, a0, false, b0, (short)0,
                                              acc, false, false);
  acc = __builtin_amdgcn_wmma_f32_16x16x4_f32(false, a1, false, b1, (short)0,
                                              acc, false, false);
  return acc;
}

// ---------------------------------------------------------------------------
// K2: value-gate GEMV  ws_c5 += c5_W @ history  (8192x8192, fp32).
// One wave owns 16 output rows x 1024-wide K-chunk = 32 cachelines of K.
// The per-line skip decision comes from a precomputed scalar bitmask
// (one dword per K-segment, forced uniform via readfirstlane), so the test
// is pure SALU and EXEC stays all-1s around the WMMAs. A nonzero line runs
// a fully unrolled dense block: 4 independent b128 A-loads + 8 WMMAs.
// ---------------------------------------------------------------------------
__global__ __launch_bounds__(256) void k2_c5(
    const float* __restrict__ c5_W, const float* __restrict__ hist,
    const unsigned* __restrict__ mask, float* __restrict__ ws_c5)
{
  const int wave   = (int)((blockIdx.x * blockDim.x + threadIdx.x) >> 5);
  const int lane   = threadIdx.x & 31;
  const int rowblk = wave / KSPLIT;
  const int kseg   = wave % KSPLIT;
  const int i0     = rowblk * 16;
  const int jb     = kseg * (VOCAB / KSPLIT);    // 1024-wide segment
  const int m      = lane & 15;
  const bool lo    = (lane < 16);
  // lanes 0-15 read cols j..j+3 ; lanes 16-31 read cols j+4..j+7
  const float* __restrict__ arow =
      c5_W + (size_t)(i0 + m) * VOCAB + (lo ? 0 : 4);

  // one mask dword covers exactly this wave's 32 lines
  const unsigned mw = __builtin_amdgcn_readfirstlane(mask[kseg]);

  v8f acc = {};
  if (mw != 0) {
    for (int l = 0; l < VOCAB / KSPLIT / LINE; ++l) {   // 32 lines
      if ((mw >> l) & 1u) {                             // scalar test
        const int j = jb + l * LINE;
        const float* __restrict__ ap = arow + j;
        const float* __restrict__ hp = hist + j;
#pragma unroll
        for (int u = 0; u < LINE; u += 8) {
          const v4f h0 = *(const v4f*)(hp + u);
          const v4f h1 = *(const v4f*)(hp + u + 4);
          const v4f a  = *(const v4f*)(ap + u);
          acc = wmma_gemv8(acc, a, h0, h1, lo);
        }
      }
    }
  }

  // every D column is identical; lane0 holds rows i0..i0+7, lane16 rows i0+8..i0+15
  if (lane == 0) {
#pragma unroll
    for (int r = 0; r < 8; ++r) atomicAdd(&ws_c5[i0 + r], acc[r]);
  } else if (lane == 16) {
#pragma unroll
    for (int r = 0; r < 8; ++r) atomicAdd(&ws_c5[i0 + 8 + r], acc[r]);
  }
}

// ---------------------------------------------------------------------------
// K3: linear_output = out_W @ h_new (+b), fused with sigmoid of the c5
// accumulator, history gating, and blended-logit write. One wave per 16
// vocab rows: 4 b128 loads + 8 WMMA f32 16x16x4 steps over K=32.
// ---------------------------------------------------------------------------
__global__ __launch_bounds__(256) void k3_logits(
    const float* __restrict__ out_W, const float* __restrict__ out_b,
    const float* __restrict__ c5_b, const float* __restrict__ hist,
    const float* __restrict__ ws, float* __restrict__ blend)
{
  const int wave = (int)((blockIdx.x * blockDim.x + threadIdx.x) >> 5);
  const int lane = threadIdx.x & 31;
  const int i0   = wave * 16;
  const int m    = lane & 15;
  const bool lo  = (lane < 16);
  const float* __restrict__ arow = out_W + (size_t)(i0 + m) * HIDDEN + (lo ? 0 : 4);
  const float* __restrict__ hvec = ws + WS_HNEW;   // h_new

  v8f acc = {};
#pragma unroll
  for (int k = 0; k < HIDDEN; k += 8) {
    const v4f h0 = *(const v4f*)(hvec + k);
    const v4f h1 = *(const v4f*)(hvec + k + 4);
    const v4f a  = *(const v4f*)(arow + k);
    acc = wmma_gemv8(acc, a, h0, h1, lo);
  }

  // lanes 0-7 cover rows i0+0..7, lanes 16-23 cover rows i0+8..15
  const int r = lane & 7;
  const bool active = ((lane & 15) < 8);
  const int row = i0 + r + ((lane >= 16) ? 8 : 0);
  float dval = 0.f;
#pragma unroll
  for (int q = 0; q < 8; ++q)
    if (q == r) dval = acc[q];
  if (active) {
    const float lin = dval + out_b[row];
    const float c5v = ws[row] + c5_b[row];
    const float val = 1.f / (1.f + expf(-c5v));
    const float h   = hist[row];
    const float res = (h != 0.f) ? 1.f : 0.f;
    blend[row] = lin * (1.f - res * val) + h * val;
  }
}

// ---------------------------------------------------------------------------
// K4: softmax over 8192 blended logits, single block of 1024 threads.
// ---------------------------------------------------------------------------
__global__ __launch_bounds__(1024) void k4_softmax(
    const float* __restrict__ blend, float* __restrict__ out)
{
  __shared__ float red[1024];
  const int t = threadIdx.x;
  float v[8];
  float mx = -INFINITY;
#pragma unroll
  for (int i = 0; i < 8; ++i) {
    v[i] = blend[t + i * 1024];
    mx = fmaxf(mx, v[i]);
  }
  red[t] = mx;
  __syncthreads();
  for (int off = 512; off > 0; off >>= 1) {
    if (t < off) red[t] = fmaxf(red[t], red[t + off]);
    __syncthreads();
  }
  mx = red[0];
  __syncthreads();
  float s = 0.f;
#pragma unroll
  for (int i = 0; i < 8; ++i) {
    v[i] = expf(v[i] - mx);
    s += v[i];
  }
  red[t] = s;
  __syncthreads();
  for (int off = 512; off > 0; off >>= 1) {
    if (t < off) red[t] += red[t + off];
    __syncthreads();
  }
  const float inv = 1.f / red[0];
#pragma unroll
  for (int i = 0; i < 8; ++i) out[t + i * 1024] = v[i] * inv;
}

// ---------------------------------------------------------------------------
extern "C" void kernel_launch(void* const* d_in, const int* in_sizes, int n_in,
                              void* d_out, int out_size, void* d_ws, size_t ws_size,
                              hipStream_t stream)
{
  const int*   ids    = (const int*)  d_in[0];
  const float* hidden = (const float*)d_in[1];
  const float* enc    = (const float*)d_in[2];
  const float* hist   = (const float*)d_in[3];
  // d_in[4] last_hidden: accepted but unused by the reference
  const float* emb    = (const float*)d_in[5];
  const float* attn_W = (const float*)d_in[6];
  const float* attn_b = (const float*)d_in[7];
  const float* comb_W = (const float*)d_in[8];
  const float* comb_b = (const float*)d_in[9];
  const float* W_ih   = (const float*)d_in[10];
  const float* W_hh   = (const float*)d_in[11];
  const float* b_ih   = (const float*)d_in[12];
  const float* b_hh   = (const float*)d_in[13];
  const float* out_W  = (const float*)d_in[14];
  const float* out_b  = (const float*)d_in[15];
  const float* c5_W   = (const float*)d_in[16];
  const float* c5_b   = (const float*)d_in[17];

  float* out      = (float*)d_out;
  float* ws       = (float*)d_ws;
  float* ws_blend = ws + WS_BLEND;
  const unsigned* ws_mask = (const unsigned*)(ws + WS_MASK);

  k1_small<<<1, 128, 0, stream>>>(ids, hidden, enc, attn_W, attn_b, comb_W, comb_b,
                                  W_ih, W_hh, b_ih, b_hh, emb, hist, out, ws);
  // 4096 waves: 512 row-blocks x 8 K-splits
  k2_c5<<<512, 256, 0, stream>>>(c5_W, hist, ws_mask, ws);
  // 512 waves: one per 16 vocab rows
  k3_logits<<<64, 256, 0, stream>>>(out_W, out_b, c5_b, hist, ws, ws_blend);
  k4_softmax<<<1, 1024, 0, stream>>>(ws_blend, out);
}